// BoundaryProximityLoss_19645180412634
// MI455X (gfx1250) — compile-verified
//
#include <hip/hip_runtime.h>

// BoundaryProximityLoss: escape-time iteration + mean reduction.
// N = 8388608 fp32 complex points, MAX_ITERS = 100, target = 30, weight = 0.1.
//
// MI455X (gfx1250, wave32) design:
//  - Pure FP32 VALU workload; hot loop compiles to VOPD dual-issue FMA pairs.
//    (No matmul structure -> WMMA semantically inapplicable; memory is ~3us
//    of streaming at 23.3 TB/s, i.e. irrelevant.)
//  - Fully predicated inner loop; per-lane bools are wave masks in SGPRs.
//    Bitwise (&,|,!) bool logic instead of short-circuit &&/|| so each float
//    condition is materialized by ONE v_cmp and negations become SGPR mask
//    ops (s_and_not1) rather than a second opposite-polarity v_cmp.
//  - Wave-uniform exit checked once per PAIR of iterations; extra predicated
//    iterations after full resolution are harmless (all updates gated on
//    'active'; active lanes have bounded z so no NaN enters live state).
//  - Iteration 0 peeled (z0=0 => z1=c) removes the "i>0" cycle guard.
//  - it starts at 100; only escape-without-cycle overwrites it. Matches
//    reference exactly: cycle => 100, unresolved => 100, cycle wins over
//    escape in the same iteration.
//  - float4 loads: one global_load_b128 per input array per thread.
//  - Deterministic two-kernel reduction: wave32 shuffle -> LDS -> per-block
//    partial in d_ws -> single-block final fold. No float atomics.

#define N_ELEMS   8388608
#define MAX_ITERS 100
#define TARGET_F  30.0f
#define WEIGHT_F  0.1f

#define BLOCK   256
#define WAVE    32
#define PER_THR 4
#define NBLOCKS (N_ELEMS / (BLOCK * PER_THR))   // 8192

__device__ __forceinline__ void esc_step(float cr, float ci,
                                         float& zr, float& zi,
                                         float& it, bool& active, float fi) {
    float zrn = zr * zr - zi * zi + cr;
    float zin = 2.0f * zr * zi + ci;

    // period-1 cycle: max(|dz_r|, |dz_i|) < 1e-6  (abs modifiers free on VALU)
    bool cyc = fmaxf(fabsf(zrn - zr), fabsf(zin - zi)) < 1e-6f;
    bool esc = (zrn * zrn + zin * zin) > 4.0f;

    // bitwise ops: one v_cmp per condition, negation as SGPR mask op
    bool resolved = cyc | esc;
    bool esc_only = active & esc & !cyc;     // cycle keeps it at 100
    it = esc_only ? fi : it;
    active = active & !resolved;

    zr = zrn;
    zi = zin;
}

__device__ __forceinline__ float escape_time(float cr, float ci) {
    // peeled i = 0: z1 = c (reference skips cycle check here)
    float zr = cr, zi = ci;
    bool  esc0   = (cr * cr + ci * ci) > 4.0f;
    bool  active = !esc0;
    float it     = esc0 ? 1.0f : (float)MAX_ITERS;

    // peeled i = 1 (so the pair loop covers i = 2..99 exactly: 49 pairs)
    esc_step(cr, ci, zr, zi, it, active, 2.0f);

    float fi = 3.0f;
    #pragma unroll 1
    for (int p = 0; p < (MAX_ITERS - 2) / 2; ++p) {
        if (__ballot(active) == 0) break;      // wave-uniform exit, 1x per pair
        esc_step(cr, ci, zr, zi, it, active, fi);
        esc_step(cr, ci, zr, zi, it, active, fi + 1.0f);
        fi += 2.0f;
    }
    return it;
}

__device__ __forceinline__ float block_reduce_sum(float v) {
    __shared__ float lds[BLOCK / WAVE];      // 8 waves per 256-thread block
    const int lane = threadIdx.x & (WAVE - 1);
    const int wid  = threadIdx.x >> 5;

    #pragma unroll
    for (int off = WAVE / 2; off > 0; off >>= 1)
        v += __shfl_down(v, off, WAVE);

    if (lane == 0) lds[wid] = v;
    __syncthreads();

    if (wid == 0) {
        v = (lane < (BLOCK / WAVE)) ? lds[lane] : 0.0f;
        #pragma unroll
        for (int off = (BLOCK / WAVE) / 2; off > 0; off >>= 1)
            v += __shfl_down(v, off, WAVE);
    }
    return v;   // valid in wave 0, lane 0
}

__global__ __launch_bounds__(BLOCK)
void escape_partial_kernel(const float4* __restrict__ cr4,
                           const float4* __restrict__ ci4,
                           float* __restrict__ partial) {
    const int i = blockIdx.x * BLOCK + threadIdx.x;   // float4 index
    const float4 cr = cr4[i];
    const float4 ci = ci4[i];

    const float inv_t = 1.0f / TARGET_F;
    float acc;
    acc  = fabsf(escape_time(cr.x, ci.x) - TARGET_F) * inv_t;
    acc += fabsf(escape_time(cr.y, ci.y) - TARGET_F) * inv_t;
    acc += fabsf(escape_time(cr.z, ci.z) - TARGET_F) * inv_t;
    acc += fabsf(escape_time(cr.w, ci.w) - TARGET_F) * inv_t;

    const float s = block_reduce_sum(acc);
    if (threadIdx.x == 0) partial[blockIdx.x] = s;
}

__global__ __launch_bounds__(BLOCK)
void final_reduce_kernel(const float* __restrict__ partial,
                         float* __restrict__ out) {
    float v = 0.0f;
    // fixed per-thread sequential order -> deterministic
    for (int i = threadIdx.x; i < NBLOCKS; i += BLOCK)
        v += partial[i];

    const float s = block_reduce_sum(v);
    if (threadIdx.x == 0)
        out[0] = s * (WEIGHT_F / (float)N_ELEMS);
}

extern "C" void kernel_launch(void* const* d_in, const int* in_sizes, int n_in,
                              void* d_out, int out_size, void* d_ws, size_t ws_size,
                              hipStream_t stream) {
    const float* c_real = (const float*)d_in[0];
    const float* c_imag = (const float*)d_in[1];
    float* out     = (float*)d_out;
    float* partial = (float*)d_ws;   // NBLOCKS floats = 32 KB

    escape_partial_kernel<<<NBLOCKS, BLOCK, 0, stream>>>(
        (const float4*)c_real, (const float4*)c_imag, partial);
    final_reduce_kernel<<<1, BLOCK, 0, stream>>>(partial, out);
}